// PredictionHead3D_13125420056652
// MI455X (gfx1250) — compile-verified
//
#include <hip/hip_runtime.h>
#include <cstddef>

#define SEG_T   0.7f
#define GAUSS_T 0.7f
#define EPSF    1e-7f
#define PI_APPROX 3.14f
#define LOG2E 1.4426950408889634f

typedef __attribute__((ext_vector_type(2))) float v2f;
typedef __attribute__((ext_vector_type(4))) float v4f;
typedef __attribute__((ext_vector_type(8))) float v8f;

// ---------------------------------------------------------------------------
// Kernel 1: build the (negated) rank-8 feature matrix such that
//   -E[p,q] = sum_k A[p,k] * B[k,q],  B[*,q] = {i^2, j^2, i*j, i, j, 1, 0, 0}
// Storage order is permuted to [f0,f1,f4,f5, f2,f3,f6,f7] so each WMMA lane
// (K-half kp) finds its four features contiguous -> one global_load_b128.
// Masked points (seg <= 0.7) get -1e30 in the constant feature: they lose
// every max, and exp(-1e30) == 0 matches the reference's where(mask, g, 0).
// ---------------------------------------------------------------------------
__global__ __launch_bounds__(256)
void prep_A_kernel(const float* __restrict__ var, const float* __restrict__ seg,
                   float* __restrict__ A) {
    const int P = 6400;
    int g = blockIdx.x * blockDim.x + threadIdx.x;     // 0 .. 2*P-1
    if (g >= 2 * P) return;
    int b = g / P, p = g % P;

    const float* vb = var + (size_t)b * 3 * P;
    float v0 = vb[p], v1 = vb[P + p], v2 = vb[2 * P + p];
    float vh = v0 * v0, vw = v1 * v1;

    float th = PI_APPROX * (1.0f / (1.0f + __expf(-v2)));
    float s = __sinf(th), c = __cosf(th);

    float ih  = 1.0f / (2.0f * vh + EPSF);
    float iw  = 1.0f / (2.0f * vw + EPSF);
    float ih4 = 1.0f / (4.0f * vh + EPSF);
    float iw4 = 1.0f / (4.0f * vw + EPSF);

    float a  = c * c * ih + s * s * iw;
    float bb = -2.0f * s * c * ih4 + 2.0f * s * c * iw4;
    float cc = s * s * ih + c * c * iw;

    float x  = (float)(p / 80), y = (float)(p % 80);
    float xp = x - EPSF, yp = y - EPSF;       // di = i - x + eps = i - xp
    float b2 = 2.0f * bb;

    float A0 = a;                                        // i^2
    float A1 = cc;                                       // j^2
    float A2 = b2;                                       // i*j
    float A3 = -(2.0f * a * xp + b2 * yp);               // i
    float A4 = -(2.0f * cc * yp + b2 * xp);              // j
    float A5 = a * xp * xp + b2 * xp * yp + cc * yp * yp;// 1
    if (!(seg[(size_t)b * P + p] > SEG_T)) A5 += 1e30f;  // mask sentinel

    // store NEGATED features, permuted: [f0,f1,f4,f5, f2,f3,f6,f7]
    float* dst = A + (size_t)g * 8;
    dst[0] = -A0; dst[1] = -A1; dst[2] = -A4; dst[3] = -A5;
    dst[4] = -A2; dst[5] = -A3; dst[6] = 0.0f; dst[7] = 0.0f;
}

// ---------------------------------------------------------------------------
// Kernel 2: one wave per 16-pixel output tile. Loop over 400 source-point
// tiles: 1x global_load_b128 + 2x V_WMMA_F32_16X16X4_F32 (K=8 as 4+4) +
// FOUR INDEPENDENT v_max3 chains over the 8 D VGPRs (raw -E values), so the
// reduction pipelines instead of serializing on one register.
// exp is monotonic: max_p exp(-E_p+eps) == exp(max_p(-E_p)+eps) -> a single
// v_exp_f32 per output pixel after the loop.
// C/D layout: VGPR e holds M=e (lanes 0-15) / M=8+e (lanes 16-31), N=lane%16;
// max over M = max over the 8 VGPRs then shfl_xor(16) across lane halves.
// ---------------------------------------------------------------------------
__global__ __launch_bounds__(256)
void splat_max_kernel(const float* __restrict__ A, float* __restrict__ out) {
    const int P = 6400;
    int lane = threadIdx.x & 31;
    int wave = threadIdx.x >> 5;
    int b    = blockIdx.y;
    int qbase = (blockIdx.x * 8 + wave) * 16;   // 400 q-tiles per batch
    int n  = lane & 15;                          // N within tile (also M for A)
    int kp = lane >> 4;                          // which K half this lane holds
    int q  = qbase + n;

    float fi = (float)(q / 80), fj = (float)(q % 80);
    v2f b0, b1;                                  // B-matrix fragments (static)
    b0.x = kp ? fi * fj : fi * fi;               // K = 0|2 : {i^2 | i*j}
    b0.y = kp ? fi      : fj * fj;               // K = 1|3 : {j^2 | i}
    b1.x = kp ? 0.0f    : fj;                    // K = 4|6 : {j   | 0}
    b1.y = kp ? 0.0f    : 1.0f;                  // K = 5|7 : {1   | 0}

    // permuted storage: lane's 4 features live at p*8 + kp*4 .. +3
    const float* ap = A + (size_t)b * P * 8 + n * 8 + kp * 4;

    // four independent running-max chains (break the VALU dep chain)
    float rm0 = -3.0e38f, rm1 = -3.0e38f, rm2 = -3.0e38f, rm3 = -3.0e38f;
    for (int pt = 0; pt < P / 16; ++pt) {
        v4f av = *(const v4f*)(ap + pt * 128);   // one b128 load per tile
        v2f a0 = { av.x, av.y };                 // K pair for step 0
        v2f a1 = { av.z, av.w };                 // K pair for step 1
        v8f acc = {};
        acc = __builtin_amdgcn_wmma_f32_16x16x4_f32(
                  false, a0, false, b0, (short)0, acc, false, false);
        acc = __builtin_amdgcn_wmma_f32_16x16x4_f32(
                  false, a1, false, b1, (short)0, acc, false, false);
        rm0 = fmaxf(rm0, fmaxf(acc[0], acc[1])); // -> v_max3_num_f32, chain 0
        rm1 = fmaxf(rm1, fmaxf(acc[2], acc[3])); // -> chain 1
        rm2 = fmaxf(rm2, fmaxf(acc[4], acc[5])); // -> chain 2
        rm3 = fmaxf(rm3, fmaxf(acc[6], acc[7])); // -> chain 3
    }
    float rmax = fmaxf(fmaxf(rm0, rm1), fmaxf(rm2, rm3));

    // merge the two lane halves (M=0..7 vs M=8..15)
    rmax = fmaxf(rmax, __shfl_xor(rmax, 16, 32));
    if (kp == 0) {
        // single exp per output pixel: g = exp(max(-E) + eps)
        float g = __builtin_amdgcn_exp2f(__builtin_fmaf(rmax, LOG2E, EPSF * LOG2E));
        out[(size_t)b * P + q] = (g >= GAUSS_T) ? g : 0.0f;
    }
}

// ---------------------------------------------------------------------------
extern "C" void kernel_launch(void* const* d_in, const int* in_sizes, int n_in,
                              void* d_out, int out_size, void* d_ws, size_t ws_size,
                              hipStream_t stream) {
    const float* var = (const float*)d_in[0];   // [2,3,80,80] f32
    const float* seg = (const float*)d_in[1];   // [2,1,80,80] f32
    float* A   = (float*)d_ws;                  // needs 2*6400*8*4 = 400 KB
    float* out = (float*)d_out;                 // [2,1,80,80] f32

    prep_A_kernel<<<dim3(50), dim3(256), 0, stream>>>(var, seg, A);
    splat_max_kernel<<<dim3(50, 2), dim3(256), 0, stream>>>(A, out);
}